// DLI_loss_1_6614249636365
// MI455X (gfx1250) — compile-verified
//
#include <hip/hip_runtime.h>
#include <hip/hip_bf16.h>
#include <stdint.h>

// ---------------------------------------------------------------------------
// DLI loss: P = E[8192,2048] @ Wpacked[2048,4]  (WMMA f32 16x16x4, async
// global->LDS double-buffered staging), then pairwise 2-class log-softmax NLL
// over k<j<len_b, mean over valid pairs.
// ---------------------------------------------------------------------------

typedef __attribute__((ext_vector_type(2))) float v2f;
typedef __attribute__((ext_vector_type(8))) float v8f;

constexpr int Dm   = 2048;   // feature dim
constexpr int Lm   = 128;    // max length
constexpr int Bm   = 64;     // batch
constexpr int ROWS = Bm * Lm;           // 8192 rows
constexpr int BK   = 32;                // K tile per stage
constexpr int WAVES_PER_BLOCK = 4;      // 128 threads
constexpr int TPB  = WAVES_PER_BLOCK * 32;
constexpr int ROWS_PER_WAVE  = 16;
constexpr int ROWS_PER_BLOCK = WAVES_PER_BLOCK * ROWS_PER_WAVE; // 64
constexpr int ET_STRIDE = BK + 4;       // 36 dwords -> 144B row stride (16B aligned)
constexpr int NT = Dm / BK;             // 64 K-tiles

__device__ __forceinline__ uint32_t lds_off(const void* p) {
    // generic pointers to LDS carry the LDS byte offset in the low 32 bits
    return (uint32_t)(uintptr_t)p;
}

__device__ __forceinline__ void async_tile(uint64_t gbase, const uint64_t* goff,
                                           uint32_t lbase, const uint32_t* loff) {
    #pragma unroll
    for (int i = 0; i < 4; ++i) {
        uint64_t ga = gbase + goff[i];
        uint32_t la = lbase + loff[i];
        asm volatile("global_load_async_to_lds_b128 %0, %1, off"
                     :: "v"(la), "v"(ga) : "memory");
    }
}

// ---------------- Kernel 1: projection GEMM via WMMA -----------------------
__global__ __launch_bounds__(TPB)
void proj_wmma(const float* __restrict__ E, const float* __restrict__ W,
               float* __restrict__ P) {
    __shared__ float Wp[Dm * 4 + 8];   // packed [k][4], +8 zero pad for N>=4 lanes
    __shared__ float Et[WAVES_PER_BLOCK][2][ROWS_PER_WAVE * ET_STRIDE]; // double buffer

    const int tid = threadIdx.x;

    // Stage packed weights: Wp[k] = {W[k,0], W[k,1], W[k+D,0], W[k+D,1]}
    for (int k = tid; k < Dm; k += TPB) {
        const float2 wl = *(const float2*)(W + 2 * (size_t)k);
        const float2 wr = *(const float2*)(W + 2 * (size_t)(k + Dm));
        float4 v; v.x = wl.x; v.y = wl.y; v.z = wr.x; v.w = wr.y;
        *(float4*)(&Wp[4 * k]) = v;
    }
    if (tid < 8) Wp[Dm * 4 + tid] = 0.0f;    // zero pad region
    __syncthreads();

    const int wave = tid >> 5;
    const int lane = tid & 31;
    const int half = lane >> 4;   // 0: lanes 0-15, 1: lanes 16-31
    const int l15  = lane & 15;
    const int rowbase = blockIdx.x * ROWS_PER_BLOCK + wave * ROWS_PER_WAVE;

    // Per-lane fixed offsets for the 16x32 tile (4 b128 transfers per lane)
    uint64_t goff[4];
    uint32_t loff[4];
    #pragma unroll
    for (int i = 0; i < 4; ++i) {
        const int t = lane + 32 * i;   // 0..127 float4 slots
        const int r = t >> 3;          // row 0..15
        const int q = t & 7;           // quad 0..7
        goff[i] = ((uint64_t)(rowbase + r) * Dm + (uint64_t)q * 4) * 4ull;
        loff[i] = (uint32_t)((r * ET_STRIDE + q * 4) * 4);
    }
    const uint64_t gbase = (uint64_t)(uintptr_t)E;
    const uint32_t lb[2] = { lds_off(&Et[wave][0][0]), lds_off(&Et[wave][1][0]) };

    v8f acc = {};   // 16x16 f32 accumulator (only N<4 meaningful)

    // prime the pipeline: tile 0 -> buffer 0
    async_tile(gbase, goff, lb[0], loff);

    int buf = 0;
    for (int t = 0; t < NT; ++t) {
        if (t + 1 < NT) {
            // prefetch next tile into the other buffer, then wait for tile t
            async_tile(gbase + (uint64_t)(t + 1) * BK * 4ull, goff, lb[buf ^ 1], loff);
            asm volatile("s_wait_asynccnt 0x4" ::: "memory");
        } else {
            asm volatile("s_wait_asynccnt 0x0" ::: "memory");
        }

        const float* et = &Et[wave][buf][0];
        const int kt = t * BK;

        #pragma unroll
        for (int kk = 0; kk < 8; ++kk) {
            // A fragment (16x4 f32): VGPR0 = K0 (lanes0-15) / K2 (lanes16-31), VGPR1 = K1/K3
            const int kb = kk * 4 + half * 2;
            v2f a = *(const v2f*)(&et[l15 * ET_STRIDE + kb]);

            // B fragment (4x16 f32): rows K0/K2 in VGPR0, K1/K3 in VGPR1, N = lane&15.
            // Lanes with N>=4 read the zero pad (index select, no exec masking).
            const int kg   = kt + kk * 4 + half * 2;
            const int bidx = (l15 < 4) ? (kg * 4 + l15) : (Dm * 4);
            v2f bv;
            bv.x = Wp[bidx];
            bv.y = Wp[bidx + 4];

            acc = __builtin_amdgcn_wmma_f32_16x16x4_f32(
                false, a, false, bv, (short)0, acc, false, false);
        }
        buf ^= 1;
    }

    // D layout: VGPR m holds row m (lanes 0-15) / row m+8 (lanes 16-31), N = lane&15
    if (l15 < 4) {
        #pragma unroll
        for (int m = 0; m < 8; ++m) {
            const int row = rowbase + m + 8 * half;
            P[(size_t)row * 4 + l15] = acc[m];
        }
    }
}

// ---------------- Kernel 2: per-batch pairwise NLL partials -----------------
__global__ __launch_bounds__(128)
void loss_partial(const float* __restrict__ P, const int* __restrict__ mask,
                  const float* __restrict__ bias,
                  float* __restrict__ psum, int* __restrict__ pcnt) {
    __shared__ float Pl[Lm * 4];
    __shared__ float red[128];
    __shared__ int   ired[128];

    const int b   = blockIdx.x;
    const int tid = threadIdx.x;

    // stage P[b] (128 x 4 floats)
    for (int i = tid; i < Lm * 4; i += 128) Pl[i] = P[(size_t)b * Lm * 4 + i];
    // length = sum(mask[b])
    ired[tid] = (tid < Lm) ? mask[b * Lm + tid] : 0;
    __syncthreads();
    for (int s = 64; s > 0; s >>= 1) {
        if (tid < s) ired[tid] += ired[tid + s];
        __syncthreads();
    }
    const int len = ired[0];
    const float b0 = bias[0], b1 = bias[1];

    float s = 0.0f;
    for (int j = tid; j < len; j += 128) {
        const float l0 = Pl[j * 4 + 0] + b0;
        const float l1 = Pl[j * 4 + 1] + b1;
        for (int k = 0; k < j; ++k) {
            const float z0 = l0 + Pl[k * 4 + 2];
            const float z1 = l1 + Pl[k * 4 + 3];
            const float dz  = fabsf(z0 - z1);
            const float lse = fmaxf(z0, z1) + log1pf(expf(-dz));
            const float zl  = (k == j - 1) ? z1 : z0;   // label 1 iff k == j-1
            s += lse - zl;                              // -log p[label]
        }
    }
    red[tid] = s;
    __syncthreads();
    for (int st = 64; st > 0; st >>= 1) {
        if (tid < st) red[tid] += red[tid + st];
        __syncthreads();
    }
    if (tid == 0) {
        psum[b] = red[0];
        pcnt[b] = (len * (len - 1)) / 2;
    }
}

// ---------------- Kernel 3: final reduction --------------------------------
__global__ __launch_bounds__(64)
void finalize(const float* __restrict__ psum, const int* __restrict__ pcnt,
              float* __restrict__ out) {
    __shared__ float fr[64];
    __shared__ int   ir[64];
    const int tid = threadIdx.x;
    fr[tid] = psum[tid];
    ir[tid] = pcnt[tid];
    __syncthreads();
    for (int s = 32; s > 0; s >>= 1) {
        if (tid < s) { fr[tid] += fr[tid + s]; ir[tid] += ir[tid + s]; }
        __syncthreads();
    }
    if (tid == 0) {
        const int cnt = ir[0] > 1 ? ir[0] : 1;
        out[0] = fr[0] / (float)cnt;
    }
}

// ---------------- Launch ----------------------------------------------------
extern "C" void kernel_launch(void* const* d_in, const int* in_sizes, int n_in,
                              void* d_out, int out_size, void* d_ws, size_t ws_size,
                              hipStream_t stream) {
    (void)in_sizes; (void)n_in; (void)out_size; (void)ws_size;

    const float* E    = (const float*)d_in[0];   // [64,128,2048] f32
    const int*   mask = (const int*)d_in[1];     // [64,128] i32
    const float* W    = (const float*)d_in[2];   // [4096,2] f32
    const float* bias = (const float*)d_in[3];   // [2] f32

    float* P    = (float*)d_ws;                                  // 8192*4 f32 = 128 KB
    float* psum = (float*)((char*)d_ws + (size_t)ROWS * 4 * sizeof(float));
    int*   pcnt = (int*)(psum + Bm);

    proj_wmma  <<<ROWS / ROWS_PER_BLOCK, TPB, 0, stream>>>(E, W, P);
    loss_partial<<<Bm, 128, 0, stream>>>(P, mask, bias, psum, pcnt);
    finalize   <<<1, 64, 0, stream>>>(psum, pcnt, (float*)d_out);
}